// DiffusionScheduler_75179107549583
// MI455X (gfx1250) — compile-verified
//
#include <hip/hip_runtime.h>
#include <hip/hip_bf16.h>
#include <math.h>

// SE(3) diffusion add-noise step for MI455X (gfx1250, wave32).
// Memory-bound (~46MB traffic); 4x4 matrix chains done on V_WMMA_F32_16X16X4_F32
// by batching 4 items per WMMA (stacked A (16x4) x stacked B (4x16); the
// diagonal 4x4 blocks of D are the per-item products, rest is discarded).

#define NUM_ITEMS (4096 * 64)      // B * HO
#define OUT_HALF  (NUM_ITEMS * 16) // floats per output tensor
#define SIGMA_R 0.05f
#define SIGMA_T 0.03f

typedef float v2f __attribute__((ext_vector_type(2)));
typedef float v8f __attribute__((ext_vector_type(8)));

// ---- per-lane SE(3) math (fp32) -------------------------------------------

__device__ __forceinline__ void se3_exp16(const float xi[6], float* H) {
    const float wx = xi[0], wy = xi[1], wz = xi[2];
    const float vx = xi[3], vy = xi[4], vz = xi[5];
    float th2  = wx * wx + wy * wy + wz * wz;
    bool small = th2 < 1e-8f;
    float th2s = small ? 1.0f : th2;
    float th   = sqrtf(th2s);
    float s, c;
    sincosf(th, &s, &c);    // shared range reduction for sin+cos
    float A  = small ? 1.0f - th2 * (1.0f / 6.0f)   : s / th;
    float Bc = small ? 0.5f - th2 * (1.0f / 24.0f)  : (1.0f - c) / th2s;
    float Cc = small ? (1.0f / 6.0f) - th2 * (1.0f / 120.0f)
                     : (th - s) / (th2s * th);
    float xx = wx * wx, yy = wy * wy, zz = wz * wz;
    float xy = wx * wy, xz = wx * wz, yz = wy * wz;
    // R = I + A*K + Bc*K^2
    float R00 = 1.0f - Bc * (yy + zz), R01 = Bc * xy - A * wz, R02 = Bc * xz + A * wy;
    float R10 = Bc * xy + A * wz, R11 = 1.0f - Bc * (xx + zz), R12 = Bc * yz - A * wx;
    float R20 = Bc * xz - A * wy, R21 = Bc * yz + A * wx, R22 = 1.0f - Bc * (xx + yy);
    // V = I + Bc*K + Cc*K^2
    float V00 = 1.0f - Cc * (yy + zz), V01 = Cc * xy - Bc * wz, V02 = Cc * xz + Bc * wy;
    float V10 = Cc * xy + Bc * wz, V11 = 1.0f - Cc * (xx + zz), V12 = Cc * yz - Bc * wx;
    float V20 = Cc * xz - Bc * wy, V21 = Cc * yz + Bc * wx, V22 = 1.0f - Cc * (xx + yy);
    float tx = V00 * vx + V01 * vy + V02 * vz;
    float ty = V10 * vx + V11 * vy + V12 * vz;
    float tz = V20 * vx + V21 * vy + V22 * vz;
    H[0] = R00; H[1] = R01; H[2]  = R02; H[3]  = tx;
    H[4] = R10; H[5] = R11; H[6]  = R12; H[7]  = ty;
    H[8] = R20; H[9] = R21; H[10] = R22; H[11] = tz;
    H[12] = 0.0f; H[13] = 0.0f; H[14] = 0.0f; H[15] = 1.0f;
}

// xi = se3_log(O^-1), computed directly from O (O^-1 = [R^T, -R^T t])
__device__ __forceinline__ void se3_log_of_inv(const float* O, float xi[6]) {
    float R00 = O[0], R01 = O[4], R02 = O[8];
    float R10 = O[1], R11 = O[5], R12 = O[9];
    float R20 = O[2], R21 = O[6], R22 = O[10];
    float tx = -(R00 * O[3] + R01 * O[7] + R02 * O[11]);
    float ty = -(R10 * O[3] + R11 * O[7] + R12 * O[11]);
    float tz = -(R20 * O[3] + R21 * O[7] + R22 * O[11]);
    float e0 = 0.5f * (R21 - R12);
    float e1 = 0.5f * (R02 - R20);
    float e2 = 0.5f * (R10 - R01);
    float cos_t = 0.5f * (R00 + R11 + R22 - 1.0f);
    cos_t = fminf(1.0f, fmaxf(-1.0f, cos_t));
    float s2 = e0 * e0 + e1 * e1 + e2 * e2;
    bool small = s2 < 1e-12f;
    float sin_s = sqrtf(small ? 1.0f : s2);
    float th = atan2f(sin_s, cos_t);
    float fac = small ? 1.0f + s2 * (1.0f / 6.0f) : th / sin_s;
    float wx = fac * e0, wy = fac * e1, wz = fac * e2;
    float omc  = small ? 1.0f : 1.0f - cos_t;
    float th2s = small ? 1.0f : th * th;
    float D = small ? (1.0f / 12.0f) + s2 * (1.0f / 720.0f)
                    : (1.0f - th * sin_s / (2.0f * omc)) / th2s;
    float xx = wx * wx, yy = wy * wy, zz = wz * wz;
    float xy = wx * wy, xz = wx * wz, yz = wy * wz;
    // Vinv = I - 0.5*K + D*K^2
    float Vi00 = 1.0f - D * (yy + zz), Vi01 = D * xy + 0.5f * wz, Vi02 = D * xz - 0.5f * wy;
    float Vi10 = D * xy - 0.5f * wz, Vi11 = 1.0f - D * (xx + zz), Vi12 = D * yz + 0.5f * wx;
    float Vi20 = D * xz + 0.5f * wy, Vi21 = D * yz - 0.5f * wx, Vi22 = 1.0f - D * (xx + yy);
    xi[0] = wx; xi[1] = wy; xi[2] = wz;
    xi[3] = Vi00 * tx + Vi01 * ty + Vi02 * tz;
    xi[4] = Vi10 * tx + Vi11 * ty + Vi12 * tz;
    xi[5] = Vi20 * tx + Vi21 * ty + Vi22 * tz;
}

// ---- kernel ----------------------------------------------------------------

__global__ __launch_bounds__(256) void
se3_diffuse_kernel(const float* __restrict__ twist,
                   const float* __restrict__ noise,
                   const float* __restrict__ alpha_bars,
                   const int* __restrict__ timesteps,
                   float* __restrict__ out) {
    // Wave-private LDS staging: [wave][item-in-wave][4x4].
    // sA: E matrices, later overwritten per-round with M1 = E*O.
    // sB: O matrices, later overwritten per-round with final product N*M1.
    // sN: H_noise matrices (also the second output).
    // sDump: scratch sink so diagonal-block extraction is branchless.
    __shared__ float sA[8][32][16];
    __shared__ float sB[8][32][16];
    __shared__ float sN[8][32][16];
    __shared__ float sDump[8][48];

    const int tid  = threadIdx.x;
    const int wave = tid >> 5;
    const int lane = tid & 31;
    const int item = blockIdx.x * 256 + tid; // grid exactly covers NUM_ITEMS
    const int b    = item >> 6;              // /HO

    // Load twist + noise (6 floats each, 8-byte aligned -> float2 loads)
    const float2* tw2 = (const float2*)(twist + (size_t)item * 6);
    const float2* nz2 = (const float2*)(noise + (size_t)item * 6);
    float2 t0 = tw2[0], t1 = tw2[1], t2 = tw2[2];
    float2 n0 = nz2[0], n1 = nz2[1], n2 = nz2[2];
    float tw[6] = {t0.x, t0.y, t1.x, t1.y, t2.x, t2.y};
    float nz[6] = {n0.x, n0.y, n1.x, n1.y, n2.x, n2.y};

    const float ab = alpha_bars[timesteps[b]];

    float O[16];
    se3_exp16(tw, O);

    float xi[6];
    se3_log_of_inv(O, xi);
    const float f1 = 1.0f - sqrtf(ab);
#pragma unroll
    for (int i = 0; i < 6; ++i) xi[i] *= f1;
    float E[16];
    se3_exp16(xi, E);

    const float f2 = sqrtf(1.0f - ab);
    float nv[6] = {f2 * SIGMA_R * nz[0], f2 * SIGMA_R * nz[1], f2 * SIGMA_R * nz[2],
                   f2 * SIGMA_T * nz[3], f2 * SIGMA_T * nz[4], f2 * SIGMA_T * nz[5]};
    float N[16];
    se3_exp16(nv, N);

#pragma unroll
    for (int i = 0; i < 16; ++i) {
        sA[wave][lane][i] = E[i];
        sB[wave][lane][i] = O[i];
        sN[wave][lane][i] = N[i];
    }
    __builtin_amdgcn_wave_barrier(); // intra-wave LDS RAW: HW DS ops are in-order

    // 8 rounds x 4 items: D = A(16x4) * B(4x16); diagonal 4x4 blocks are the
    // per-item products. A row M = item (M/4), row (M%4); B col N likewise.
    const int half = lane >> 4;     // K-half select (A: K 0,1 vs 2,3; B rows)
    const int q    = lane & 15;     // M for A-layout, N for B-layout
    const int sub  = q & 3;
    const int grp  = q >> 2;

    // Diagonal extraction: only lanes with grp>>1 == half carry block-diagonal
    // data; they hold d[4*(grp&1) .. +3]. Others sink into sDump (no branches).
    const bool act = (grp >> 1) == half;
    const bool hi  = (grp & 1) != 0;
    float* const dumpBase = &sDump[wave][lane];

#pragma unroll
    for (int r = 0; r < 8; ++r) {
        const int it = 4 * r + grp;

        v2f a, bb;
        a.x  = sA[wave][it][sub * 4 + 2 * half];        // E[item][M%4][K]
        a.y  = sA[wave][it][sub * 4 + 2 * half + 1];
        bb.x = sB[wave][it][(2 * half) * 4 + sub];      // O[item][K][N%4]
        bb.y = sB[wave][it][(2 * half + 1) * 4 + sub];

        v8f c = {};
        v8f d = __builtin_amdgcn_wmma_f32_16x16x4_f32(
            false, a, false, bb, (short)0, c, false, false);

        // M1 = E*O diagonal blocks -> this round's E slots (branchless).
        {
            float* p = act ? &sA[wave][it][sub] : dumpBase;
            p[0]  = hi ? d[4] : d[0];
            p[4]  = hi ? d[5] : d[1];
            p[8]  = hi ? d[6] : d[2];
            p[12] = hi ? d[7] : d[3];
        }
        __builtin_amdgcn_wave_barrier();

        v2f an, bm;
        an.x = sN[wave][it][sub * 4 + 2 * half];        // N as A operand
        an.y = sN[wave][it][sub * 4 + 2 * half + 1];
        bm.x = sA[wave][it][(2 * half) * 4 + sub];      // M1 as B operand
        bm.y = sA[wave][it][(2 * half + 1) * 4 + sub];

        v8f d2 = __builtin_amdgcn_wmma_f32_16x16x4_f32(
            false, an, false, bm, (short)0, c, false, false);

        // Final product N*E*O -> this round's O slots (flat layout == output).
        {
            float* p = act ? &sB[wave][it][sub] : dumpBase;
            p[0]  = hi ? d2[4] : d2[0];
            p[4]  = hi ? d2[5] : d2[1];
            p[8]  = hi ? d2[6] : d2[2];
            p[12] = hi ? d2[7] : d2[3];
        }
        __builtin_amdgcn_wave_barrier();
    }

    // Coalesced 128-bit stores, wave-private region -> global.
    // flat layout [wave][item][16] matches item-major output exactly.
    const float4* s1 = (const float4*)(&sB[wave][0][0]);
    const float4* s2 = (const float4*)(&sN[wave][0][0]);
    float4* o1 = ((float4*)out) + (size_t)blockIdx.x * 1024 + wave * 128;
    float4* o2 = ((float4*)(out + OUT_HALF)) + (size_t)blockIdx.x * 1024 + wave * 128;
#pragma unroll
    for (int j = 0; j < 4; ++j) {
        o1[lane + j * 32] = s1[lane + j * 32];
        o2[lane + j * 32] = s2[lane + j * 32];
    }
}

extern "C" void kernel_launch(void* const* d_in, const int* in_sizes, int n_in,
                              void* d_out, int out_size, void* d_ws, size_t ws_size,
                              hipStream_t stream) {
    const float* twist      = (const float*)d_in[0];
    const float* noise      = (const float*)d_in[1];
    const float* alpha_bars = (const float*)d_in[2];
    const int*   timesteps  = (const int*)d_in[3];
    float*       out        = (float*)d_out;

    const int blocks = NUM_ITEMS / 256; // 1024, exact coverage (EXEC all-ones)
    hipLaunchKernelGGL(se3_diffuse_kernel, dim3(blocks), dim3(256), 0, stream,
                       twist, noise, alpha_bars, timesteps, out);
}